// FactorizedSpectralConv2d_72962904425080
// MI455X (gfx1250) — compile-verified
//
#include <hip/hip_runtime.h>
#include <hip/hip_bf16.h>

// FNO FactorizedSpectralConv2d for gfx1250 (MI455X).
// B=16, CIN=COUT=64, H=W=128, MH=MW=32, super_res=1.
// Five WMMA-GEMM stages through workspace (all intermediates L2-resident).
// Twiddles come from a 128-entry LDS LUT (all angles are 2*pi*p/128).
// Stage A stages x tiles into LDS via async-load-to-LDS when available.

typedef __attribute__((ext_vector_type(2))) float v2f;
typedef __attribute__((ext_vector_type(8))) float v8f;

#define NB    16
#define NCI   64
#define NCO   64
#define NH    128
#define NW    128
#define NM    32
#define C128  0.049087385212340514f   /* 2*pi/128 */
#define INV_HW (1.0f / 16384.0f)

#if __has_builtin(__builtin_amdgcn_global_load_async_to_lds_b128) && \
    __has_builtin(__builtin_amdgcn_s_wait_asynccnt)
#define USE_ASYNC_LDS 1
#else
#define USE_ASYNC_LDS 0
#endif

#if USE_ASYNC_LDS
// Builtin signature (from hipcc diagnostic): param0 is
// 'int __attribute__((vector_size(16))) __device__ *'  (= int4, AS1).
typedef int v4i_vs __attribute__((vector_size(16)));
typedef __attribute__((address_space(1))) v4i_vs as1_v4i;
typedef __attribute__((address_space(3))) v4i_vs as3_v4i;
#endif

__device__ __forceinline__ v8f wmma_f32(v2f a, v2f b, v8f c) {
  // D = A(16x4,f32) * B(4x16,f32) + C(16x16,f32)
  return __builtin_amdgcn_wmma_f32_16x16x4_f32(
      false, a, false, b, (short)0, c, false, false);
}

// Per-block twiddle LUT: twd[p] = {cos(2*pi*p/128), sin(2*pi*p/128)}
__device__ __forceinline__ void tw_init(float2* twd) {
  if (threadIdx.x < 128) {
    float s, c; __sincosf(C128 * (float)threadIdx.x, &s, &c);
    float2 v; v.x = c; v.y = s;
    twd[threadIdx.x] = v;
  }
}

// -------------------------------------------------------------------------
// Stage A: pruned rfft along W.  x[B,CIN,H,W] -> SA[row][kw]{re,im}, kw<32
// GEMM: M = B*CIN*H rows (131072), K = 128, N = 64 (re0..31, im0..31)
// 128 threads/block = 4 waves; one 16-row M-tile per wave.
// -------------------------------------------------------------------------
#define XS_PITCH 132   /* padded row pitch (floats): 16B aligned, bank-spread */

__global__ void __launch_bounds__(128)
fno_rfft_w(const float* __restrict__ x, float* __restrict__ SA) {
  __shared__ float2 twd[128];
#if USE_ASYNC_LDS
  __shared__ float xs[4][16 * XS_PITCH];
#endif
  const int lane  = threadIdx.x & 31;
  const int wave  = threadIdx.x >> 5;
  const int tile  = blockIdx.x * 4 + wave;      // 8192 tiles
  const int row0  = tile * 16;
  const int nloc  = lane & 15;
  const int khalf = lane >> 4;

  tw_init(twd);

#if USE_ASYNC_LDS
  {
    // Stage this wave's 16x128 x tile into LDS: 16 coalesced async B128 ops.
    const int rrow = lane >> 1;                 // 0..15
    const int half = lane & 1;                  // which 16B of a 32B chunk
    const size_t   gbase = (size_t)(x + (row0 + rrow) * NW + half * 4);
    const unsigned lbase = (unsigned)(size_t)(&xs[wave][rrow * XS_PITCH + half * 4]);
#pragma unroll
    for (int j = 0; j < 16; ++j) {
      __builtin_amdgcn_global_load_async_to_lds_b128(
          (as1_v4i*)(gbase + (size_t)j * 32), (as3_v4i*)(lbase + j * 32), 0, 0);
    }
  }
  __syncthreads();                              // twd ready
  __builtin_amdgcn_s_wait_asynccnt(0);          // x tile ready (wave-private)
#else
  __syncthreads();
  const float* __restrict__ xrow = x + (row0 + nloc) * NW;
#endif

  v8f acc[4] = {v8f{}, v8f{}, v8f{}, v8f{}};
  for (int kb = 0; kb < NW; kb += 4) {
    const int k0 = kb + 2 * khalf;
    v2f a;
#if USE_ASYNC_LDS
    const float2 xa = *(const float2*)&xs[wave][nloc * XS_PITCH + k0];
    a[0] = xa.x; a[1] = xa.y;
#else
    a[0] = xrow[k0]; a[1] = xrow[k0 + 1];
#endif
#pragma unroll
    for (int t = 0; t < 4; ++t) {
      const int kw = (t & 1) * 16 + nloc;       // output frequency 0..31
      v2f b;
#pragma unroll
      for (int v = 0; v < 2; ++v) {
        const float2 tt = twd[((k0 + v) * kw) & 127];
        b[v] = (t < 2) ? tt.x : -tt.y;          // re: cos, im: -sin
      }
      acc[t] = wmma_f32(a, b, acc[t]);
    }
  }
#pragma unroll
  for (int t = 0; t < 4; ++t) {
    const int kw = (t & 1) * 16 + nloc;
    const int part = t >> 1;                    // 0=re 1=im
#pragma unroll
    for (int g = 0; g < 8; ++g) {
      const int r = row0 + g + 8 * khalf;
      SA[(r * NM + kw) * 2 + part] = acc[t][g];
    }
  }
}

// -------------------------------------------------------------------------
// Stage B: column FFT (len 128), keep rows 0..31 and 96..127.
// SA[b,ci,h,kw] -> SB[b,ci,r,kw], r=0..63 maps to freq (r<32? r : 96+r-32)
// GEMM: M = columns (b,ci,kw) = 32768, K = 128 (complex), N = 64
// -------------------------------------------------------------------------
__global__ void __launch_bounds__(256)
fno_fft_h(const float* __restrict__ SA_, float* __restrict__ SB_) {
  __shared__ float2 twd[128];
  const float2* __restrict__ SA = (const float2*)SA_;
  float2* __restrict__ SB = (float2*)SB_;
  const int lane  = threadIdx.x & 31;
  const int wave  = threadIdx.x >> 5;
  const int tile  = blockIdx.x * 8 + wave;      // 2048 tiles
  const int cbase = tile * 16;
  const int nloc  = lane & 15;
  const int khalf = lane >> 4;

  tw_init(twd);
  __syncthreads();

  const int cA  = cbase + nloc;                 // column for A fragment
  const int kwA = cA & 31, bciA = cA >> 5;

  v8f yr[4] = {v8f{}, v8f{}, v8f{}, v8f{}};
  v8f yi[4] = {v8f{}, v8f{}, v8f{}, v8f{}};

  for (int kb = 0; kb < NH; kb += 4) {
    const int h0 = kb + 2 * khalf;
    const float2 z0 = SA[(bciA * NH + h0) * NM + kwA];
    const float2 z1 = SA[(bciA * NH + h0 + 1) * NM + kwA];
    v2f ar; ar[0] = z0.x; ar[1] = z1.x;
    v2f ai; ai[0] = z0.y; ai[1] = z1.y;
#pragma unroll
    for (int t = 0; t < 4; ++t) {
      const int r = t * 16 + nloc;
      const int f = (r < 32) ? r : (96 + (r - 32));
      v2f dr, di, dn;
#pragma unroll
      for (int v = 0; v < 2; ++v) {
        const float2 tt = twd[((h0 + v) * f) & 127];   // e^{-i theta}
        dr[v] = tt.x; di[v] = -tt.y; dn[v] = tt.y;
      }
      yr[t] = wmma_f32(ar, dr, yr[t]);
      yr[t] = wmma_f32(ai, dn, yr[t]);   // -Ai*Di
      yi[t] = wmma_f32(ar, di, yi[t]);
      yi[t] = wmma_f32(ai, dr, yi[t]);
    }
  }
#pragma unroll
  for (int t = 0; t < 4; ++t) {
    const int r = t * 16 + nloc;
#pragma unroll
    for (int g = 0; g < 8; ++g) {
      const int c  = cbase + g + 8 * khalf;
      const int kw = c & 31, bci = c >> 5;
      float2 v2; v2.x = yr[t][g]; v2.y = yi[t][g];
      SB[(bci * 64 + r) * NM + kw] = v2;
    }
  }
}

// -------------------------------------------------------------------------
// Stage C: per-mode channel contraction (the einsum).
// For mode (s,x,y): out[b,o] = sum_i SB[b,i,r,y] * W[s,i,o,x,y] (complex)
// GEMM: M = batch 16, K = CIN 64, N = COUT 64.  One wave per mode (2048).
// -------------------------------------------------------------------------
__global__ void __launch_bounds__(256)
fno_modemul(const float* __restrict__ SB_, const float* __restrict__ Wt,
            float* __restrict__ SC_) {
  const float2* __restrict__ SB = (const float2*)SB_;
  float2* __restrict__ SC = (float2*)SC_;
  const int lane  = threadIdx.x & 31;
  const int wave  = threadIdx.x >> 5;
  const int tile  = blockIdx.x * 8 + wave;      // 2048 modes
  const int s = tile >> 10;
  const int xm = (tile >> 5) & 31;
  const int ym = tile & 31;
  const int r = s * 32 + xm;
  const int nloc  = lane & 15;
  const int khalf = lane >> 4;
  const int b = nloc;                           // A-fragment M index

  v8f orr[4] = {v8f{}, v8f{}, v8f{}, v8f{}};
  v8f oii[4] = {v8f{}, v8f{}, v8f{}, v8f{}};

  for (int kb = 0; kb < NCI; kb += 4) {
    const int ci0 = kb + 2 * khalf;
    // Prefetch next ci-chunk's weight lines (each line = 32 y values,
    // shared via L2 by the 32 waves handling the other y of this (s,x)).
    if (kb + 4 < NCI) {
      const int pb = (((s * NCI + (ci0 + 4)) * NCO + nloc) << 10) + xm * 32 + ym;
      __builtin_prefetch(&Wt[pb], 0, 1);
      __builtin_prefetch(&Wt[pb + 8388608], 0, 1);
    }
    const float2 z0 = SB[((b * NCI + ci0) * 64 + r) * NM + ym];
    const float2 z1 = SB[((b * NCI + ci0 + 1) * 64 + r) * NM + ym];
    v2f ar; ar[0] = z0.x; ar[1] = z1.x;
    v2f ai; ai[0] = z0.y; ai[1] = z1.y;
#pragma unroll
    for (int t = 0; t < 4; ++t) {
      const int o = t * 16 + nloc;
      v2f wr, wi, wn;
#pragma unroll
      for (int v = 0; v < 2; ++v) {
        const int ci = ci0 + v;
        // weight[reim][s][ci][o][x][y], reim stride = 2*64*64*1024
        const int base = (((s * NCI + ci) * NCO + o) << 10) + xm * 32 + ym;
        wr[v] = Wt[base];
        const float im = Wt[base + 8388608];
        wi[v] = im; wn[v] = -im;
      }
      orr[t] = wmma_f32(ar, wr, orr[t]);
      orr[t] = wmma_f32(ai, wn, orr[t]);
      oii[t] = wmma_f32(ar, wi, oii[t]);
      oii[t] = wmma_f32(ai, wr, oii[t]);
    }
  }
#pragma unroll
  for (int t = 0; t < 4; ++t) {
    const int o = t * 16 + nloc;
#pragma unroll
    for (int g = 0; g < 8; ++g) {
      const int bb = g + 8 * khalf;
      float2 v2; v2.x = orr[t][g]; v2.y = oii[t][g];
      SC[((bb * NCO + o) * 64 + r) * NM + ym] = v2;
    }
  }
}

// -------------------------------------------------------------------------
// Stage D1: inverse column FFT: 64 retained rows -> 128 spatial rows.
// SC[b,co,r,kw] -> SD[b,co,h,kw].  GEMM: M = cols (b,co,kw), K=64, N=128.
// -------------------------------------------------------------------------
__global__ void __launch_bounds__(256)
fno_ifft_h(const float* __restrict__ SC_, float* __restrict__ SD_) {
  __shared__ float2 twd[128];
  const float2* __restrict__ SC = (const float2*)SC_;
  float2* __restrict__ SD = (float2*)SD_;
  const int lane  = threadIdx.x & 31;
  const int wave  = threadIdx.x >> 5;
  const int tile  = blockIdx.x * 8 + wave;      // 2048 tiles
  const int cbase = tile * 16;
  const int nloc  = lane & 15;
  const int khalf = lane >> 4;

  tw_init(twd);
  __syncthreads();

  const int cA  = cbase + nloc;
  const int kwA = cA & 31, bcoA = cA >> 5;

  for (int nt = 0; nt < 8; ++nt) {              // h tiles
    v8f zr = v8f{}, zi = v8f{};
    const int h = nt * 16 + nloc;
    for (int kb = 0; kb < 64; kb += 4) {
      const int r0 = kb + 2 * khalf;
      const float2 y0 = SC[(bcoA * 64 + r0) * NM + kwA];
      const float2 y1 = SC[(bcoA * 64 + r0 + 1) * NM + kwA];
      v2f ar; ar[0] = y0.x; ar[1] = y1.x;
      v2f ai; ai[0] = y0.y; ai[1] = y1.y;
      v2f dr, di, dn;
#pragma unroll
      for (int v = 0; v < 2; ++v) {
        const int rr = r0 + v;
        const int f = (rr < 32) ? rr : (96 + (rr - 32));
        const float2 tt = twd[(h * f) & 127];   // e^{+i theta}
        dr[v] = tt.x; di[v] = tt.y; dn[v] = -tt.y;
      }
      zr = wmma_f32(ar, dr, zr);
      zr = wmma_f32(ai, dn, zr);
      zi = wmma_f32(ar, di, zi);
      zi = wmma_f32(ai, dr, zi);
    }
#pragma unroll
    for (int g = 0; g < 8; ++g) {
      const int c  = cbase + g + 8 * khalf;
      const int kw = c & 31, bco = c >> 5;
      float2 v2; v2.x = zr[g]; v2.y = zi[g];
      SD[(bco * NH + h) * NM + kw] = v2;
    }
  }
}

// -------------------------------------------------------------------------
// Stage D2: inverse rfft along W (32 nonzero freqs, Hermitian folded),
// 1/(H*W) normalization and bias.  GEMM: M = B*CO*H rows, K=64, N=128.
// y[n] = inv*(Xr0 + 2*sum_{k=1..31}(Xr[k]cos - Xi[k]sin))
// -------------------------------------------------------------------------
__global__ void __launch_bounds__(256)
fno_irfft_w(const float* __restrict__ SD_, const float* __restrict__ bias,
            float* __restrict__ out) {
  __shared__ float2 twd[128];
  const float2* __restrict__ SD = (const float2*)SD_;
  const int lane  = threadIdx.x & 31;
  const int wave  = threadIdx.x >> 5;
  const int tile  = blockIdx.x * 8 + wave;      // 8192 tiles
  const int row0  = tile * 16;
  const int nloc  = lane & 15;
  const int khalf = lane >> 4;
  const int rowA  = row0 + nloc;

  tw_init(twd);
  __syncthreads();

  v8f acc[8] = {v8f{}, v8f{}, v8f{}, v8f{}, v8f{}, v8f{}, v8f{}, v8f{}};

  for (int kb = 0; kb < 64; kb += 4) {
    const int k0 = kb + 2 * khalf;
    v2f a;
#pragma unroll
    for (int v = 0; v < 2; ++v) {
      const int k = k0 + v;
      const float2 z = SD[rowA * NM + ((k < 32) ? k : (k - 32))];
      a[v] = (k < 32) ? z.x : z.y;
    }
#pragma unroll
    for (int nt = 0; nt < 8; ++nt) {
      const int w = nt * 16 + nloc;
      v2f bf;
#pragma unroll
      for (int v = 0; v < 2; ++v) {
        const int k = k0 + v;
        const int kk = (k < 32) ? k : (k - 32);
        const float ck = (kk == 0) ? INV_HW : (2.0f * INV_HW);
        const float2 tt = twd[(kk * w) & 127];
        bf[v] = (k < 32) ? (ck * tt.x) : (-ck * tt.y);
      }
      acc[nt] = wmma_f32(a, bf, acc[nt]);
    }
  }
#pragma unroll
  for (int nt = 0; nt < 8; ++nt) {
    const int w = nt * 16 + nloc;
#pragma unroll
    for (int g = 0; g < 8; ++g) {
      const int row = row0 + g + 8 * khalf;
      const int co = (row >> 7) & (NCO - 1);
      out[row * NW + w] = acc[nt][g] + bias[co];
    }
  }
}

// -------------------------------------------------------------------------
extern "C" void kernel_launch(void* const* d_in, const int* in_sizes, int n_in,
                              void* d_out, int out_size, void* d_ws, size_t ws_size,
                              hipStream_t stream) {
  const float* x    = (const float*)d_in[0];   // [16,64,128,128]
  const float* wt   = (const float*)d_in[1];   // [2,2,64,64,32,32]
  const float* bias = (const float*)d_in[2];   // [64]
  // d_in[3] = super_res == 1 (ignored)
  float* out = (float*)d_out;

  float* ws = (float*)d_ws;
  float* SA = ws;                  // [B*CIN*H, 32] complex   = 33.5 MB
  float* SB = ws + 8388608;        // [B*CIN, 64, 32] complex = 16.8 MB
  float* SC = ws + 12582912;       // [B*CO, 64, 32] complex  = 16.8 MB
  float* SD = ws + 16777216;       // [B*CO, 128, 32] complex = 33.5 MB
  (void)in_sizes; (void)n_in; (void)out_size; (void)ws_size;

  fno_rfft_w <<<2048, 128, 0, stream>>>(x, SA);
  fno_fft_h  <<<256,  256, 0, stream>>>(SA, SB);
  fno_modemul<<<256,  256, 0, stream>>>(SB, wt, SC);
  fno_ifft_h <<<256,  256, 0, stream>>>(SC, SD);
  fno_irfft_w<<<1024, 256, 0, stream>>>(SD, bias, out);
}